// BaseModel_91096256348406
// MI455X (gfx1250) — compile-verified
//
#include <hip/hip_runtime.h>
#include <hip/hip_bf16.h>

typedef __attribute__((ext_vector_type(2))) float v2f;
typedef __attribute__((ext_vector_type(8))) float v8f;
typedef __attribute__((ext_vector_type(4))) float v4f;
typedef __attribute__((ext_vector_type(4))) int   v4i;
typedef __attribute__((ext_vector_type(2))) float v2fv;

#define N_VERTS 4194304
#define H 512

// ---------------------------------------------------------------------------
// prep: build padded encoder input x (1536 floats, zero padded) and the
// decoder conditioning vector dec_in (1536 floats, zero padded):
//   [0:66)  prev_joint_positions = 0
//   [66:69) prev_root_velocity   = 0
//   [69:135)  skel (flattened)
//   [135:263) geo_emb
//   [263:775) h_enc1  (written later)
//   [775:1287) h_enc2 (written later)
// ---------------------------------------------------------------------------
__global__ void prep_kernel(const float* __restrict__ rot,
                            const float* __restrict__ vel,
                            const float* __restrict__ skel,
                            const float* __restrict__ geo,
                            float* __restrict__ ws_x,
                            float* __restrict__ ws_di) {
  int t = blockIdx.x * blockDim.x + threadIdx.x;
  if (t < 1536) {
    float xv = 0.0f;
    if (t < 66)       xv = rot[t];
    else if (t < 69)  xv = vel[t - 66];
    ws_x[t] = xv;

    float dv = 0.0f;
    if (t >= 69 && t < 135)        dv = skel[t - 69];
    else if (t >= 135 && t < 263)  dv = geo[t - 135];
    ws_di[t] = dv;
  }
}

// ---------------------------------------------------------------------------
// GEMV via V_WMMA_F32_16X16X4_F32 (full fp32 precision).
// One wave (blockDim=32) computes 16 consecutive output rows:
//   out[r..r+15] = W[r..r+15, :] @ x      (bias added later)
// A tile layout (16x4 f32): lanes 0-15 -> M=lane, K={0,1}; lanes 16-31 ->
// M=lane-16, K={2,3}.  B (4x16) carries x replicated in every column:
// lanes 0-15 hold rows K={0,1}, lanes 16-31 rows K={2,3} -> uniform per half.
// D column 0 lands in lane 0 (M=0..7) and lane 16 (M=8..15).
// ---------------------------------------------------------------------------
__global__ void gemv_wmma_kernel(const float* __restrict__ W,
                                 const float* __restrict__ x,
                                 float* __restrict__ out,
                                 int IN) {
  const int wave = blockIdx.x;                 // 96 tiles for 1536 rows
  const int lane = threadIdx.x;                // 0..31
  const int mrow = lane & 15;
  const int koff = (lane >> 4) * 2;            // 0 (low half) or 2 (high half)
  const float* __restrict__ Wrow = W + (size_t)(wave * 16 + mrow) * (size_t)IN;

  v8f c0 = {}; v8f c1 = {};
  const int nfull = IN >> 2;                   // full K=4 chunks

  int i = 0;
  for (; i + 2 <= nfull; i += 2) {
    int k0 = (i << 2) + koff;
    v2f a0, b0;
    a0.x = Wrow[k0];     a0.y = Wrow[k0 + 1];
    b0.x = x[k0];        b0.y = x[k0 + 1];
    c0 = __builtin_amdgcn_wmma_f32_16x16x4_f32(false, a0, false, b0,
                                               (short)0, c0, false, false);
    int k1 = k0 + 4;
    v2f a1, b1;
    a1.x = Wrow[k1];     a1.y = Wrow[k1 + 1];
    b1.x = x[k1];        b1.y = x[k1 + 1];
    c1 = __builtin_amdgcn_wmma_f32_16x16x4_f32(false, a1, false, b1,
                                               (short)0, c1, false, false);
  }
  if (i < nfull) {
    int k0 = (i << 2) + koff;
    v2f a0, b0;
    a0.x = Wrow[k0];     a0.y = Wrow[k0 + 1];
    b0.x = x[k0];        b0.y = x[k0 + 1];
    c0 = __builtin_amdgcn_wmma_f32_16x16x4_f32(false, a0, false, b0,
                                               (short)0, c0, false, false);
    ++i;
  }
  if (IN & 3) {                                // guarded tail chunk
    int k = (nfull << 2) + koff;
    v2f a, b;
    a.x = (k     < IN) ? Wrow[k]     : 0.0f;
    a.y = (k + 1 < IN) ? Wrow[k + 1] : 0.0f;
    b.x = (k     < IN) ? x[k]        : 0.0f;
    b.y = (k + 1 < IN) ? x[k + 1]    : 0.0f;
    c0 = __builtin_amdgcn_wmma_f32_16x16x4_f32(false, a, false, b,
                                               (short)0, c0, false, false);
  }

  if ((lane & 15) == 0) {                      // lanes 0 and 16: column N=0
    int rbase = wave * 16 + (lane >> 4) * 8;
#pragma unroll
    for (int m = 0; m < 8; ++m) out[rbase + m] = c0[m] + c1[m];
  }
}

// ---------------------------------------------------------------------------
// GRU cell epilogue with h_prev == 0 (so gh == bhh):
//   r = sigmoid(gi_r + bih_r + bhh_r)
//   z = sigmoid(gi_z + bih_z + bhh_z)
//   n = tanh(gi_n + bih_n + r * bhh_n)
//   h = (1 - z) * n
// gi holds the raw dot products (no bias).
// ---------------------------------------------------------------------------
__device__ __forceinline__ float sigmoidf_(float v) {
  return 1.0f / (1.0f + expf(-v));
}

__global__ void gru_pointwise_kernel(const float* __restrict__ gi,
                                     const float* __restrict__ bih,
                                     const float* __restrict__ bhh,
                                     float* __restrict__ dst0,
                                     float* __restrict__ dst1) {
  int j = blockIdx.x * blockDim.x + threadIdx.x;
  if (j >= H) return;
  float r = sigmoidf_(gi[j]         + bih[j]         + bhh[j]);
  float z = sigmoidf_(gi[H + j]     + bih[H + j]     + bhh[H + j]);
  float n = tanhf(    gi[2 * H + j] + bih[2 * H + j] + r * bhh[2 * H + j]);
  float h = (1.0f - z) * n;
  dst0[j] = h;
  if (dst1 != nullptr) dst1[j] = h;
}

// ---------------------------------------------------------------------------
// Heads: local_rot = rotW @ h + rotb (66), root_vel = velW @ h + velb (3),
// global_positions[t] = local_rot[t] * root_vel[t % 3]  -> d_out[0:66)
// local_rot also stashed in workspace for the skinning pass.
// ---------------------------------------------------------------------------
__global__ void heads_kernel(const float* __restrict__ h,
                             const float* __restrict__ rotW,
                             const float* __restrict__ rotb,
                             const float* __restrict__ velW,
                             const float* __restrict__ velb,
                             float* __restrict__ gpos,
                             float* __restrict__ rot_ws) {
  __shared__ float rot_s[66];
  __shared__ float vel_s[3];
  int t = threadIdx.x;
  for (int row = t; row < 69; row += blockDim.x) {
    const float* Wr = (row < 66) ? (rotW + (size_t)row * H)
                                 : (velW + (size_t)(row - 66) * H);
    float acc = 0.0f;
    for (int k = 0; k < H; ++k) acc = fmaf(Wr[k], h[k], acc);
    acc += (row < 66) ? rotb[row] : velb[row - 66];
    if (row < 66) rot_s[row] = acc; else vel_s[row - 66] = acc;
  }
  __syncthreads();
  if (t < 66) {
    rot_ws[t] = rot_s[t];
    gpos[t]   = rot_s[t] * vel_s[t % 3];
  }
}

// ---------------------------------------------------------------------------
// Skinning (HBM-bound, ~235 MB single pass):
//   out[n] = vertices[n] * sum(w) + sum_k w[k] * offsets[idx[k]]
// 4 vertices per thread: positions read as 3x B128 (48 B, 16 B-aligned),
// weights/indices as 4x B128 each, output written as 6x B64 stores
// (outp base is only 8-byte aligned: d_out + 66 floats = 264 B).
// All big streams use non-temporal hints (single-use data).
// offsets (22x3) staged in LDS.
// ---------------------------------------------------------------------------
__global__ void skin_kernel(const float* __restrict__ verts,
                            const int*   __restrict__ sidx,
                            const float* __restrict__ sw,
                            const float* __restrict__ rot_ws,
                            float* __restrict__ outp) {
  __shared__ float offs[66];
  if (threadIdx.x < 66) offs[threadIdx.x] = rot_ws[threadIdx.x];
  __syncthreads();

  const int t = blockIdx.x * blockDim.x + threadIdx.x;   // vertex-quad id
  // 4 vertices per thread -> 12 position floats as 3x v4f (NT B128 loads)
  const v4f* __restrict__ v4 = (const v4f*)verts;
  v4f va = __builtin_nontemporal_load(&v4[3 * t + 0]);
  v4f vb = __builtin_nontemporal_load(&v4[3 * t + 1]);
  v4f vc = __builtin_nontemporal_load(&v4[3 * t + 2]);

  float v[12] = {va.x, va.y, va.z, va.w,
                 vb.x, vb.y, vb.z, vb.w,
                 vc.x, vc.y, vc.z, vc.w};
  float o[12];

  const v4f* __restrict__ w4  = (const v4f*)sw;
  const v4i* __restrict__ id4 = (const v4i*)sidx;

#pragma unroll
  for (int q = 0; q < 4; ++q) {
    const int n = 4 * t + q;
    const v4f w  = __builtin_nontemporal_load(&w4[n]);
    const v4i id = __builtin_nontemporal_load(&id4[n]);
    const float wsum = w.x + w.y + w.z + w.w;
    const int b0 = 3 * id.x, b1 = 3 * id.y, b2 = 3 * id.z, b3 = 3 * id.w;
#pragma unroll
    for (int d = 0; d < 3; ++d) {
      float g = w.x * offs[b0 + d] + w.y * offs[b1 + d] +
                w.z * offs[b2 + d] + w.w * offs[b3 + d];
      o[3 * q + d] = fmaf(v[3 * q + d], wsum, g);
    }
  }

  // 12 floats out -> 6x v2fv (B64) NT stores, 8-byte aligned base
  v2fv* __restrict__ o2 = (v2fv*)outp;
#pragma unroll
  for (int p = 0; p < 6; ++p) {
    v2fv st;
    st.x = o[2 * p];
    st.y = o[2 * p + 1];
    __builtin_nontemporal_store(st, &o2[6 * t + p]);
  }
}

// ---------------------------------------------------------------------------
// Launch. Input order = setup_inputs() dict insertion order, params flattened
// recursively: enc[0](Wih,Whh,bih,bhh), enc[1](...), dec[0](...), dec[1](...),
// rotW, rotb, velW, velb, skel, geo_emb.  Whh matrices are unused (h0 == 0).
// ---------------------------------------------------------------------------
extern "C" void kernel_launch(void* const* d_in, const int* in_sizes, int n_in,
                              void* d_out, int out_size, void* d_ws, size_t ws_size,
                              hipStream_t stream) {
  const float* rotation = (const float*)d_in[0];
  const float* velocity = (const float*)d_in[1];
  const float* vertices = (const float*)d_in[2];
  const int*   skin_idx = (const int*)  d_in[3];
  const float* skin_w   = (const float*)d_in[4];

  const float* enc0_Wih = (const float*)d_in[5];
  const float* enc0_bih = (const float*)d_in[7];
  const float* enc0_bhh = (const float*)d_in[8];
  const float* enc1_Wih = (const float*)d_in[9];
  const float* enc1_bih = (const float*)d_in[11];
  const float* enc1_bhh = (const float*)d_in[12];
  const float* dec0_Wih = (const float*)d_in[13];
  const float* dec0_bih = (const float*)d_in[15];
  const float* dec0_bhh = (const float*)d_in[16];
  const float* dec1_Wih = (const float*)d_in[17];
  const float* dec1_bih = (const float*)d_in[19];
  const float* dec1_bhh = (const float*)d_in[20];
  const float* rotW     = (const float*)d_in[21];
  const float* rotb     = (const float*)d_in[22];
  const float* velW     = (const float*)d_in[23];
  const float* velb     = (const float*)d_in[24];
  const float* skel     = (const float*)d_in[25];
  const float* geo      = (const float*)d_in[26];

  float* ws     = (float*)d_ws;
  float* ws_x   = ws;          // [0, 1536)   current GEMV input (zero padded)
  float* ws_gi  = ws + 1536;   // [1536,3072) raw gate dot products
  float* ws_di  = ws + 3072;   // [3072,4608) dec_in (1287 used, zero padded)
  float* ws_rot = ws + 4608;   // [4608,4674) local_rot for skinning
  float* outv   = (float*)d_out;

  prep_kernel<<<6, 256, 0, stream>>>(rotation, velocity, skel, geo, ws_x, ws_di);

  // encoder layer 1 (IN=69)
  gemv_wmma_kernel<<<96, 32, 0, stream>>>(enc0_Wih, ws_x, ws_gi, 69);
  gru_pointwise_kernel<<<2, 256, 0, stream>>>(ws_gi, enc0_bih, enc0_bhh,
                                              ws_x, ws_di + 263);
  // encoder layer 2 (IN=512)
  gemv_wmma_kernel<<<96, 32, 0, stream>>>(enc1_Wih, ws_x, ws_gi, 512);
  gru_pointwise_kernel<<<2, 256, 0, stream>>>(ws_gi, enc1_bih, enc1_bhh,
                                              ws_di + 775, nullptr);
  // decoder layer 1 (IN=1287)
  gemv_wmma_kernel<<<96, 32, 0, stream>>>(dec0_Wih, ws_di, ws_gi, 1287);
  gru_pointwise_kernel<<<2, 256, 0, stream>>>(ws_gi, dec0_bih, dec0_bhh,
                                              ws_x, nullptr);
  // decoder layer 2 (IN=512)
  gemv_wmma_kernel<<<96, 32, 0, stream>>>(dec1_Wih, ws_x, ws_gi, 512);
  gru_pointwise_kernel<<<2, 256, 0, stream>>>(ws_gi, dec1_bih, dec1_bhh,
                                              ws_x, nullptr);

  // output heads + global_positions -> d_out[0:66)
  heads_kernel<<<1, 128, 0, stream>>>(ws_x, rotW, rotb, velW, velb,
                                      outv, ws_rot);

  // skinning -> d_out[66 : 66 + 3*N), 4 vertices per thread
  skin_kernel<<<N_VERTS / (256 * 4), 256, 0, stream>>>(vertices, skin_idx,
                                                       skin_w, ws_rot,
                                                       outv + 66);
}